// Repeat_decoder_add_43636867727568
// MI455X (gfx1250) — compile-verified
//
#include <hip/hip_runtime.h>

#define B_ 512
#define S_ 200
#define D_ 64
#define V_ 100000
#define PAD_GAP_ 180.0f
#define LOG_SQRT_2PI_ 0.9189385332046727f

typedef float v2f __attribute__((ext_vector_type(2)));
typedef float v8f __attribute__((ext_vector_type(8)));

__global__ void zero_kernel(float* __restrict__ out, long n) {
    long tid    = (long)blockIdx.x * blockDim.x + threadIdx.x;
    long stride = (long)gridDim.x * blockDim.x;
    long n4 = n >> 2;
    float4* o4 = (float4*)out;
    float4 z; z.x = 0.f; z.y = 0.f; z.z = 0.f; z.w = 0.f;
    for (long i = tid; i < n4; i += stride) o4[i] = z;
    for (long i = (n4 << 2) + tid; i < n; i += stride) out[i] = 0.f;
}

__device__ __forceinline__ float gauss_pdf(float x, float mu, float sg) {
    float z = (x - mu) / sg;
    return __expf(-0.5f * z * z - __logf(sg) - LOG_SQRT_2PI_);
}

__global__ __launch_bounds__(256) void fused_kernel(
    const float* __restrict__ x, const int* __restrict__ x_ids,
    const float* __restrict__ time_gap, const int* __restrict__ cla,
    const float* __restrict__ Wq, const float* __restrict__ bq,
    const float* __restrict__ Wk, const float* __restrict__ bk,
    const float* __restrict__ wv, const float* __restrict__ bv,
    const float* __restrict__ theta, float* __restrict__ out)
{
    __shared__ float s_wq[D_ * D_];   // Wq staged, row-major [k][n]
    __shared__ float s_kq[D_];        // k_b + bk + bq
    __shared__ float s_wv[D_];
    __shared__ float s_theta[17];
    __shared__ float s_scores[208];
    __shared__ float s_red[256];
    __shared__ int   s_id[2 * S_];    // 399 used
    __shared__ float s_val[2 * S_];

    const int tid = threadIdx.x;
    const int b   = blockIdx.x;
    const float* xb = x + (size_t)b * S_ * D_;

    // ---- Phase 0: stage weights, compute k_b + bk + bq ----
    for (int i = tid; i < D_ * D_; i += 256) s_wq[i] = Wq[i];
    for (int i = tid; i < D_; i += 256)      s_wv[i] = wv[i];
    for (int i = tid; i < 17; i += 256)      s_theta[i] = theta[i];
    if (tid < D_) {
        float acc = bk[tid] + bq[tid];
        for (int k = 0; k < D_; ++k) acc += xb[k] * Wk[k * D_ + tid];
        s_kq[tid] = acc;
    }
    __syncthreads();

    // ---- Phase 1: WMMA GEMM q = x_b @ Wq, fused tanh(q+k)·wv -> scores ----
    const int wave = tid >> 5;
    const int lane = tid & 31;
    const int half = lane >> 4;   // 0: lanes 0-15, 1: lanes 16-31
    const int l16  = lane & 15;
    const int nchunks = (S_ + 15) / 16;   // 13

    for (int chunk = wave; chunk < nchunks; chunk += 8) {
        const int s0   = chunk * 16;
        const int srow = s0 + l16;                 // A row handled by this lane
        // Clamp instead of zero: WMMA output row m depends only on A row m,
        // and rows >= S_ are never written back, so loading a duplicate of the
        // last valid row is safe and keeps the loads unconditional (b64, no
        // EXEC save/restore per load).
        const int srowc = (srow < S_) ? srow : (S_ - 1);
        const float* arow = xb + (size_t)srowc * D_ + 2 * half;

        v8f acc0 = {}, acc1 = {}, acc2 = {}, acc3 = {};
#pragma unroll
        for (int kk = 0; kk < 16; ++kk) {
            const int k0 = kk * 4 + half * 2;      // this lane's K pair
            // A frag: aligned 8-byte load (k0 even, row 256B-aligned)
            v2f a = *(const v2f*)(arow + kk * 4);
            // B frags (4x16 KxN) from LDS: VGPR0 = K{k0}, VGPR1 = K{k0+1}, N = l16
            v2f b0, b1, b2, b3;
            b0.x = s_wq[k0 * D_ +  0 + l16]; b0.y = s_wq[(k0 + 1) * D_ +  0 + l16];
            b1.x = s_wq[k0 * D_ + 16 + l16]; b1.y = s_wq[(k0 + 1) * D_ + 16 + l16];
            b2.x = s_wq[k0 * D_ + 32 + l16]; b2.y = s_wq[(k0 + 1) * D_ + 32 + l16];
            b3.x = s_wq[k0 * D_ + 48 + l16]; b3.y = s_wq[(k0 + 1) * D_ + 48 + l16];
            acc0 = __builtin_amdgcn_wmma_f32_16x16x4_f32(false, a, false, b0, (short)0, acc0, false, false);
            acc1 = __builtin_amdgcn_wmma_f32_16x16x4_f32(false, a, false, b1, (short)0, acc1, false, false);
            acc2 = __builtin_amdgcn_wmma_f32_16x16x4_f32(false, a, false, b2, (short)0, acc2, false, false);
            acc3 = __builtin_amdgcn_wmma_f32_16x16x4_f32(false, a, false, b3, (short)0, acc3, false, false);
        }

        // Epilogue: C layout -> VGPR i holds row (i + 8*half), col = nt*16 + l16
#pragma unroll
        for (int i = 0; i < 8; ++i) {
            float p = 0.f;
            {
                int n = 0 + l16;  p += tanhf(acc0[i] + s_kq[n]) * s_wv[n];
            }
            {
                int n = 16 + l16; p += tanhf(acc1[i] + s_kq[n]) * s_wv[n];
            }
            {
                int n = 32 + l16; p += tanhf(acc2[i] + s_kq[n]) * s_wv[n];
            }
            {
                int n = 48 + l16; p += tanhf(acc3[i] + s_kq[n]) * s_wv[n];
            }
            // reduce across the 16-lane group (masks < 16 stay within group)
            for (int m = 1; m < 16; m <<= 1) p += __shfl_xor(p, m, 32);
            const int row = s0 + i + half * 8;
            if (l16 == 0 && row < S_) s_scores[row] = p;
        }
    }
    __syncthreads();

    // ---- Phase 2: masked softmax over S ----
    const float bvv = bv[0];
    int   myid = 0;
    float my_score = -__builtin_inff();
    if (tid < S_) {
        myid = x_ids[(size_t)b * S_ + tid];
        const bool m = (myid != 0) && (myid != 1);
        my_score = m ? (s_scores[tid] + bvv) : -__builtin_inff();
    }
    s_red[tid] = my_score;
    __syncthreads();
    for (int off = 128; off > 0; off >>= 1) {
        if (tid < off) s_red[tid] = fmaxf(s_red[tid], s_red[tid + off]);
        __syncthreads();
    }
    const float mx = s_red[0];
    __syncthreads();
    const float ev = (tid < S_) ? __expf(my_score - mx) : 0.f;
    s_red[tid] = ev;
    __syncthreads();
    for (int off = 128; off > 0; off >>= 1) {
        if (tid < off) s_red[tid] += s_red[tid + off];
        __syncthreads();
    }
    const float denom = s_red[0];
    __syncthreads();

    // ---- Phase 3: build scatter list (deterministic) ----
    const float w = s_theta[6];
    if (tid < S_) {
        s_id[tid]  = myid;
        s_val[tid] = (1.0f - w) * (ev / denom);
    }
    if (tid < S_ - 1) {
        const float gap = time_gap[(size_t)b * (S_ - 1) + tid];
        const int   cl  = cla[(size_t)b * (S_ - 1) + tid];
        const float c0 = (cl == 0) ? gap : PAD_GAP_;
        const float c1 = (cl == 1) ? gap : PAD_GAP_;
        const float c2 = (cl == 2) ? gap : PAD_GAP_;
        float g = s_theta[0] * gauss_pdf(c0, s_theta[7],  s_theta[8])
                + s_theta[1] * gauss_pdf(c1, s_theta[9],  s_theta[10])
                + s_theta[2] * gauss_pdf(c1, s_theta[11], s_theta[12])
                + s_theta[3] * gauss_pdf(c2, s_theta[13], s_theta[14])
                + s_theta[4] * __powf(c2, s_theta[15]);
        s_id[S_ + tid]  = x_ids[(size_t)b * S_ + tid + 1];
        s_val[S_ + tid] = w * g;
    }
    __syncthreads();

    // Collision-resolved scatter: entry owner = lowest index with this id,
    // sums all matches in fixed ascending order -> deterministic, no atomics.
    const int NE = 2 * S_ - 1;   // 399
    for (int e0 = tid; e0 < NE; e0 += 256) {
        const int id = s_id[e0];
        bool owner = true;
        for (int j = 0; j < e0; ++j) {
            if (s_id[j] == id) { owner = false; break; }
        }
        if (owner) {
            float sum = s_val[e0];
            for (int j = e0 + 1; j < NE; ++j) {
                if (s_id[j] == id) sum += s_val[j];
            }
            out[(size_t)b * V_ + id] = sum;
        }
    }
}

extern "C" void kernel_launch(void* const* d_in, const int* in_sizes, int n_in,
                              void* d_out, int out_size, void* d_ws, size_t ws_size,
                              hipStream_t stream) {
    const float* x        = (const float*)d_in[0];
    const int*   x_ids    = (const int*)d_in[1];
    const float* time_gap = (const float*)d_in[2];
    const int*   cla      = (const int*)d_in[3];
    const float* Wq       = (const float*)d_in[4];
    const float* bq       = (const float*)d_in[5];
    const float* Wk       = (const float*)d_in[6];
    const float* bk       = (const float*)d_in[7];
    const float* wv       = (const float*)d_in[8];
    const float* bv       = (const float*)d_in[9];
    const float* theta    = (const float*)d_in[10];
    float* out = (float*)d_out;

    // Zero the (B,V) output: this is the bandwidth-dominant step (~205 MB).
    zero_kernel<<<2048, 256, 0, stream>>>(out, (long)out_size);
    // One block per batch row: WMMA GEMM + softmax + deterministic scatter.
    fused_kernel<<<B_, 256, 0, stream>>>(x, x_ids, time_gap, cla,
                                         Wq, bq, Wk, bk, wv, bv, theta, out);
}